// GGNNTorch_39573828665872
// MI455X (gfx1250) — compile-verified
//
#include <hip/hip_runtime.h>
#include <math.h>

typedef __attribute__((ext_vector_type(2))) float v2f;
typedef __attribute__((ext_vector_type(8))) float v8f;

#define DDIM   128
#define KQ     10
#define WAVES  8                 // waves per block (kernel 1)
#define SLICEB 8                 // blocks along base dimension
#define NSLICE (WAVES * SLICEB)  // 64 wave-slices over the base set

__device__ __forceinline__ void topk_insert(float key, int cand,
                                            float (&h)[KQ], int (&ci)[KQ]) {
  // h[] kept ascending; h[KQ-1] is the current worst of the best-10.
  if (key < h[KQ - 1]) {
    h[KQ - 1] = key; ci[KQ - 1] = cand;
#pragma unroll
    for (int j = KQ - 1; j > 0; --j) {
      if (h[j] < h[j - 1]) {
        float tf = h[j]; h[j] = h[j - 1]; h[j - 1] = tf;
        int   ti = ci[j]; ci[j] = ci[j - 1]; ci[j - 1] = ti;
      }
    }
  }
}

// ---------------------------------------------------------------------------
// Kernel 1: per (16-query tile, base wave-slice) partial top-10 using
// V_WMMA_F32_16X16X4_F32, software-pipelined over base tiles.
// ---------------------------------------------------------------------------
__global__ __launch_bounds__(256) void knn_partial_kernel(
    const float* __restrict__ base, const float* __restrict__ query,
    float* __restrict__ pKey, int* __restrict__ pIdx,
    int N, int numTiles, int tilesPerSlice)
{
  __shared__ float lds[WAVES][256];           // one 16x16 key tile per wave
  const int lane  = threadIdx.x & 31;
  const int wave  = threadIdx.x >> 5;
  const int half  = lane >> 4;                // which K-half this lane loads
  const int row   = lane & 15;                // M (for A) / N (for B) index
  const int qTile = blockIdx.x;
  const int slice = blockIdx.y * WAVES + wave;

  // ---- Preload A fragments: 16 queries x 128, resident in 64 VGPRs -------
  v2f aF[32];
  {
    const float* qrow = query + (size_t)(qTile * 16 + row) * DDIM + 2 * half;
#pragma unroll
    for (int k4 = 0; k4 < 32; ++k4)
      aF[k4] = *(const v2f*)(qrow + 4 * k4);
  }

  float h[KQ]; int ci[KQ];
#pragma unroll
  for (int j = 0; j < KQ; ++j) { h[j] = 3.0e38f; ci[j] = -1; }

  const int t0   = slice * tilesPerSlice;
  const int tEnd = min(t0 + tilesPerSlice, numTiles);
  const int nT   = tEnd - t0;

  // Issue all 32 b64 loads of one base tile into a live register array so
  // they stay in flight together (no per-load s_wait_loadcnt serialization).
  auto loadTile = [&](int t, v2f (&bF)[32]) {
    int nrow = t * 16 + row; if (nrow >= N) nrow = N - 1;      // safe clamp
    const float* brow = base + (size_t)nrow * DDIM + 2 * half;
#pragma unroll
    for (int k4 = 0; k4 < 32; ++k4)
      bF[k4] = *(const v2f*)(brow + 4 * k4);
    // Warm L2 a few tiles ahead (global_prefetch_b8).
    int tp = t + 4;
    if (tp < numTiles) {
      int np = tp * 16 + row; if (np >= N) np = N - 1;
      __builtin_prefetch(base + (size_t)np * DDIM + 2 * half, 0, 2);
    }
  };

  auto computeTile = [&](int t, const v2f (&bF)[32]) {
    v8f c = {0.f, 0.f, 0.f, 0.f, 0.f, 0.f, 0.f, 0.f};
    float bs2 = 0.f;                          // this lane's half of ||b||^2
#pragma unroll
    for (int k4 = 0; k4 < 32; ++k4) {
      bs2 = fmaf(bF[k4].x, bF[k4].x, bs2);
      bs2 = fmaf(bF[k4].y, bF[k4].y, bs2);
      c = __builtin_amdgcn_wmma_f32_16x16x4_f32(
              false, aF[k4], false, bF[k4], (short)0, c, false, false);
    }
    // Lanes r and r+16 hold complementary K-halves of the SAME base row.
    const float bsq = bs2 + __shfl_xor(bs2, 16, 32);

    // keys: key(m, n=row) = ||b_n||^2 - 2*dot(q_m, b_n), m = i + 8*half
#pragma unroll
    for (int i = 0; i < 8; ++i)
      lds[wave][(i + 8 * half) * 16 + row] = fmaf(-2.0f, c[i], bsq);

    __builtin_amdgcn_wave_barrier();          // LDS is in-order within a wave;
    asm volatile("" ::: "memory");            // just stop compiler reordering

    if (lane < 16) {                          // lane L owns query m = L
#pragma unroll
      for (int n = 0; n < 16; ++n) {
        const float key  = lds[wave][lane * 16 + n];
        const int   cand = t * 16 + n;
        if (cand < N) topk_insert(key, cand, h, ci);
      }
    }
    __builtin_amdgcn_wave_barrier();
    asm volatile("" ::: "memory");
  };

  if (nT > 0) {
    v2f b0[32], b1[32];
    loadTile(t0, b0);
    int j = 0;
    for (; j + 2 <= nT; j += 2) {             // 2-stage ping-pong pipeline
      loadTile(t0 + j + 1, b1);
      computeTile(t0 + j, b0);
      if (j + 2 < nT) loadTile(t0 + j + 2, b0);
      computeTile(t0 + j + 1, b1);
    }
    if (j < nT) computeTile(t0 + j, b0);      // odd tail
  }

  if (lane < 16) {
    const size_t o = ((size_t)(qTile * NSLICE + slice) * 16 + lane) * KQ;
#pragma unroll
    for (int j = 0; j < KQ; ++j) { pKey[o + j] = h[j]; pIdx[o + j] = ci[j]; }
  }
}

// ---------------------------------------------------------------------------
// Kernel 2: one wave per query merges the 64 partial lists, adds ||q||^2,
// emits int32 indices then fp32 distances.
// ---------------------------------------------------------------------------
__global__ __launch_bounds__(256) void knn_reduce_kernel(
    const float* __restrict__ query,
    const float* __restrict__ pKey, const int* __restrict__ pIdx,
    int* __restrict__ outIdx, float* __restrict__ outDist, int Q)
{
  __shared__ float kbuf[WAVES][32 * KQ];
  __shared__ int   ibuf[WAVES][32 * KQ];
  const int lane = threadIdx.x & 31;
  const int wave = threadIdx.x >> 5;
  const int q    = blockIdx.x * WAVES + wave;
  if (q >= Q) return;                         // uniform per wave
  const int qTile = q >> 4, m = q & 15;

  float h[KQ]; int ci[KQ];
#pragma unroll
  for (int j = 0; j < KQ; ++j) { h[j] = 3.0e38f; ci[j] = -1; }

  for (int s = lane; s < NSLICE; s += 32) {   // 2 partial lists per lane
    const size_t o = ((size_t)(qTile * NSLICE + s) * 16 + m) * KQ;
#pragma unroll
    for (int j = 0; j < KQ; ++j) {
      const float key = pKey[o + j]; const int cand = pIdx[o + j];
      if (cand >= 0) topk_insert(key, cand, h, ci);
    }
  }
#pragma unroll
  for (int j = 0; j < KQ; ++j) {
    kbuf[wave][lane * KQ + j] = h[j];
    ibuf[wave][lane * KQ + j] = ci[j];
  }
  __builtin_amdgcn_wave_barrier();
  asm volatile("" ::: "memory");

  // ||q||^2 via butterfly reduction (4 elements per lane).
  float p = 0.f;
  const float* qr = query + (size_t)q * DDIM + lane * 4;
#pragma unroll
  for (int e = 0; e < 4; ++e) p = fmaf(qr[e], qr[e], p);
#pragma unroll
  for (int off = 16; off >= 1; off >>= 1) p += __shfl_xor(p, off, 32);

  if (lane == 0) {
#pragma unroll
    for (int j = 0; j < KQ; ++j) { h[j] = 3.0e38f; ci[j] = -1; }
    for (int n = 0; n < 32 * KQ; ++n) {
      const float key = kbuf[wave][n]; const int cand = ibuf[wave][n];
      if (cand >= 0) topk_insert(key, cand, h, ci);
    }
    for (int j = 0; j < KQ; ++j) {
      outIdx[(size_t)q * KQ + j]  = ci[j];
      outDist[(size_t)q * KQ + j] = sqrtf(fmaxf(p + h[j], 0.f));
    }
  }
}

// ---------------------------------------------------------------------------
extern "C" void kernel_launch(void* const* d_in, const int* in_sizes, int n_in,
                              void* d_out, int out_size, void* d_ws, size_t ws_size,
                              hipStream_t stream) {
  const float* base  = (const float*)d_in[0];   // [N,128] fp32
  const float* query = (const float*)d_in[1];   // [Q,128] fp32
  const int N = in_sizes[0] / DDIM;
  const int Q = in_sizes[1] / DDIM;

  const int numTiles      = (N + 15) / 16;
  const int tilesPerSlice = (numTiles + NSLICE - 1) / NSLICE;
  const int qTiles        = (Q + 15) / 16;      // Q=1024 -> 64

  float* pKey = (float*)d_ws;
  const size_t nPart = (size_t)qTiles * NSLICE * 16 * KQ;
  int*   pIdx = (int*)((char*)d_ws + nPart * sizeof(float));

  int*   outIdx  = (int*)d_out;                 // int32 indices [Q,10]
  float* outDist = (float*)d_out + (size_t)Q * KQ;  // fp32 distances [Q,10]

  dim3 g1(qTiles, SLICEB), b1(256);
  knn_partial_kernel<<<g1, b1, 0, stream>>>(base, query, pKey, pIdx,
                                            N, numTiles, tilesPerSlice);

  dim3 g2((Q + WAVES - 1) / WAVES), b2(256);
  knn_reduce_kernel<<<g2, b2, 0, stream>>>(query, pKey, pIdx,
                                           outIdx, outDist, Q);
}